// GraphConvolution_24927990186546
// MI455X (gfx1250) — compile-verified
//
#include <hip/hip_runtime.h>

#define DIN  512
#define DOUT 128

typedef __attribute__((ext_vector_type(2))) float v2f;
typedef __attribute__((ext_vector_type(8))) float v8f;

// ---------------------------------------------------------------------------
// Phase 1: h[N, DOUT] = x[N, DIN] @ W[DIN, DOUT], exact fp32 via
// V_WMMA_F32_16X16X4_F32. One wave owns a 16-row strip across all 128 output
// columns (8 accumulators) so the A fragment is reused 8x per K-step.
// ---------------------------------------------------------------------------
__global__ __launch_bounds__(256) void gcn_gemm_wmma(
    const float* __restrict__ x, const float* __restrict__ W,
    float* __restrict__ h, int nRows, int nTilesM) {
  const int lane  = threadIdx.x & 31;
  const int tileM = blockIdx.x * 8 + (threadIdx.x >> 5);
  if (tileM >= nTilesM) return;                  // uniform per-wave exit

  const int m   = lane & 15;                     // row-within-tile / col-within-tile
  const int kq  = (lane >> 4) << 1;              // 0 for lanes 0-15, 2 for 16-31
  int row       = tileM * 16 + m;
  const int rowClamped = (row < nRows) ? row : (nRows - 1);
  const float* __restrict__ xr = x + (long long)rowClamped * DIN;

  v8f acc[8] = {};                               // 8 x (16x16 f32 tile)

  for (int k = 0; k < DIN; k += 4) {
    // A fragment: 16x4 tile of x. float2 load (k+kq is even -> 8B aligned).
    v2f a = *(const v2f*)(xr + k + kq);
    const float* __restrict__ wk0 = W + (k + kq) * DOUT;
    const float* __restrict__ wk1 = wk0 + DOUT;
#pragma unroll
    for (int t = 0; t < 8; ++t) {
      const int n = t * 16 + m;
      v2f b;
      b.x = wk0[n];                              // B: K rows striped over 2 VGPRs
      b.y = wk1[n];
      acc[t] = __builtin_amdgcn_wmma_f32_16x16x4_f32(
          /*neg_a=*/false, a, /*neg_b=*/false, b,
          /*c_mod=*/(short)0, acc[t], /*reuse_a=*/false, /*reuse_b=*/false);
    }
  }

  // C/D layout: VGPR j, lane l -> M = j + 8*(l>=16), N = l & 15
  const int mHalf = (lane >> 4) * 8;
#pragma unroll
  for (int t = 0; t < 8; ++t) {
#pragma unroll
    for (int j = 0; j < 8; ++j) {
      const int rr = tileM * 16 + mHalf + j;
      if (rr < nRows) h[(long long)rr * DOUT + t * 16 + m] = acc[t][j];
    }
  }
}

// ---------------------------------------------------------------------------
// Phase 2: out = relu(segment_sum(adj_vals * h[adj_cols], adj_rows)).
// adj_rows is sorted -> one wave per output row, binary-search its edge range,
// gather h rows as coalesced float4 (L2-resident: 51MB < 192MB), fuse ReLU.
// Deterministic: no atomics, every output element written exactly once.
// ---------------------------------------------------------------------------
__global__ __launch_bounds__(256) void gcn_spmm_relu(
    const float* __restrict__ h,
    const long long* __restrict__ rows,
    const long long* __restrict__ cols,
    const float* __restrict__ vals,
    long long E, int nRows, float* __restrict__ out) {
  const int lane = threadIdx.x & 31;
  const long long r = (long long)blockIdx.x * 8 + (threadIdx.x >> 5);
  if (r >= nRows) return;                        // uniform per-wave exit

  // lower_bound(rows, r) and lower_bound(rows, r+1) — uniform across the wave.
  long long lo = 0, hi = E;
  while (lo < hi) { long long mid = (lo + hi) >> 1; if (rows[mid] <  r) lo = mid + 1; else hi = mid; }
  const long long start = lo;
  hi = E;
  while (lo < hi) { long long mid = (lo + hi) >> 1; if (rows[mid] <= r) lo = mid + 1; else hi = mid; }
  const long long end = lo;

  const float4* __restrict__ h4 = (const float4*)h;  // 4 features per lane
  float4 acc = make_float4(0.f, 0.f, 0.f, 0.f);

  for (long long e = start; e < end; ++e) {
    const long long c = cols[e];
    const float     v = vals[e];
    const float4   hv = h4[c * (DOUT / 4) + lane];   // global_load_b128, L2 hit
    acc.x = fmaf(v, hv.x, acc.x);
    acc.y = fmaf(v, hv.y, acc.y);
    acc.z = fmaf(v, hv.z, acc.z);
    acc.w = fmaf(v, hv.w, acc.w);
  }

  float4 o;
  o.x = fmaxf(acc.x, 0.f);
  o.y = fmaxf(acc.y, 0.f);
  o.z = fmaxf(acc.z, 0.f);
  o.w = fmaxf(acc.w, 0.f);
  ((float4*)out)[r * (DOUT / 4) + lane] = o;
}

// ---------------------------------------------------------------------------
extern "C" void kernel_launch(void* const* d_in, const int* in_sizes, int n_in,
                              void* d_out, int out_size, void* d_ws, size_t ws_size,
                              hipStream_t stream) {
  const float*     x    = (const float*)d_in[0];
  const float*     W    = (const float*)d_in[1];
  const long long* rows = (const long long*)d_in[2];
  const long long* cols = (const long long*)d_in[3];
  const float*     vals = (const float*)d_in[4];

  const int       nRows = in_sizes[0] / DIN;   // N = 100000
  const long long E     = in_sizes[2];         // 3.2M edges

  float* h   = (float*)d_ws;                   // N*DOUT*4 = 51.2 MB scratch
  float* out = (float*)d_out;

  const int nTilesM    = (nRows + 15) / 16;    // 6250
  const int gemmBlocks = (nTilesM + 7) / 8;    // 8 waves / block
  gcn_gemm_wmma<<<gemmBlocks, 256, 0, stream>>>(x, W, h, nRows, nTilesM);

  const int rowBlocks = (nRows + 7) / 8;       // one wave per output row
  gcn_spmm_relu<<<rowBlocks, 256, 0, stream>>>(h, rows, cols, vals, E, nRows, out);
}